// RealNVP_EnVI_24953759989995
// MI455X (gfx1250) — compile-verified
//
#include <hip/hip_runtime.h>
#include <hip/hip_bf16.h>
#include <math.h>

typedef float v2f __attribute__((ext_vector_type(2)));
typedef float v8f __attribute__((ext_vector_type(8)));

#define T_STEPS 50
#define B_SZ 64
#define NMC_ 100
#define DX_ 16
#define DY_ 16
#define M_ 128
#define HID_ 64
#define NPTS (NMC_*B_SZ)           /* 6400 */
#define PVAR_ 0.0025f
#define EVAR_ 0.01f
#define JIT_ 1e-4f
#define LOG2PI_ 1.8378770664093453f

__device__ __forceinline__ v8f wmma4(v2f a, v2f b, v8f c) {
  // D = A(16x4,f32) * B(4x16,f32) + C(16x16,f32)
  return __builtin_amdgcn_wmma_f32_16x16x4_f32(false, a, false, b, (short)0, c, false, false);
}
__device__ __forceinline__ float sigf(float x)  { return 1.f/(1.f+__expf(-x)); }
__device__ __forceinline__ float leaky(float x) { return x > 0.f ? x : 0.01f*x; }
__device__ __forceinline__ int   tri(int i, int j) { return ((i*(i+1))>>1) + j; }

// Gaspari-Cohn taper entry
__device__ __forceinline__ float gc_taper(int di, int dj) {
  int diff = di > dj ? di - dj : dj - di;
  int w = 16 - diff; if (w < diff) diff = w;
  float z = (float)diff / 5.0f;
  float z2 = z*z, z3 = z2*z, z4 = z2*z2, z5 = z4*z;
  if (z < 1.f) return 1.f - (5.f/3.f)*z2 + (5.f/8.f)*z3 + 0.5f*z4 - 0.25f*z5;
  if (z < 2.f) return 4.f - 5.f*z + (5.f/3.f)*z2 + (5.f/8.f)*z3 - 0.5f*z4 + (1.f/12.f)*z5 - 2.f/(3.f*z);
  return 0.f;
}

// ---------------------------------------------------------------- zero accums
__global__ void k_zero(float* accum) {
  if (threadIdx.x < 4) accum[threadIdx.x] = 0.f;
}

// ---------------------------------------------------------------- LSTM + q(x0)
__global__ __launch_bounds__(256) void k_lstm(
    const float* __restrict__ obs, const float* __restrict__ Wih,
    const float* __restrict__ Whh, const float* __restrict__ b_lstm,
    const float* __restrict__ Wm, const float* __restrict__ bm,
    const float* __restrict__ Wv, const float* __restrict__ bv,
    float* __restrict__ m0, float* __restrict__ v0, float* __restrict__ accum)
{
  __shared__ float hbuf[B_SZ*HID_];     // 64x64
  __shared__ float ybuf[B_SZ*DY_];      // 64x16
  __shared__ float gbuf[32*256];        // half-batch gates
  const int tid  = threadIdx.x;
  const int wave = tid >> 5;
  const int lane = tid & 31;
  const int lo = lane & 15, hi = lane >> 4;

  for (int i = tid; i < B_SZ*HID_; i += 256) hbuf[i] = 0.f;
  float creg[2][8];
  #pragma unroll
  for (int hh = 0; hh < 2; ++hh)
    #pragma unroll
    for (int s = 0; s < 8; ++s) creg[hh][s] = 0.f;
  __syncthreads();

  for (int t = 0; t < T_STEPS; ++t) {
    for (int i = tid; i < B_SZ*DY_; i += 256) {
      int b = i >> 4, dy = i & 15;
      ybuf[i] = obs[(b*T_STEPS + t)*DY_ + dy];
    }
    __syncthreads();
    for (int half = 0; half < 2; ++half) {
      // g = y_t @ Wih^T + h @ Whh^T + b  (rows = 32 batch rows of this half)
      for (int tt = wave; tt < 32; tt += 8) {
        int rt = tt >> 4, ct = tt & 15;
        int brow = half*32 + rt*16;
        v8f c = {0.f,0.f,0.f,0.f,0.f,0.f,0.f,0.f};
        #pragma unroll
        for (int kk = 0; kk < 4; ++kk) {           // K over y (16)
          int kb = kk*4 + 2*hi;
          v2f a, bb;
          a.x  = ybuf[(brow+lo)*DY_ + kb];   a.y  = ybuf[(brow+lo)*DY_ + kb + 1];
          bb.x = Wih[(ct*16+lo)*DY_ + kb];   bb.y = Wih[(ct*16+lo)*DY_ + kb + 1];
          c = wmma4(a, bb, c);
        }
        #pragma unroll
        for (int kk = 0; kk < 16; ++kk) {          // K over h (64)
          int kb = kk*4 + 2*hi;
          v2f a, bb;
          a.x  = hbuf[(brow+lo)*HID_ + kb];  a.y  = hbuf[(brow+lo)*HID_ + kb + 1];
          bb.x = Whh[(ct*16+lo)*HID_ + kb];  bb.y = Whh[(ct*16+lo)*HID_ + kb + 1];
          c = wmma4(a, bb, c);
        }
        float bias = b_lstm[ct*16 + lo];
        #pragma unroll
        for (int v = 0; v < 8; ++v)
          gbuf[(rt*16 + v + 8*hi)*256 + ct*16 + lo] = c[v] + bias;
      }
      __syncthreads();
      // gate update: 32x64 = 2048 (b,j) pairs, 8 per thread, c-state in registers
      #pragma unroll
      for (int s = 0; s < 8; ++s) {
        int idx = tid*8 + s;
        int bl = idx >> 6, j = idx & 63;
        float gi = gbuf[bl*256 + j];
        float gf = gbuf[bl*256 + j + 64];
        float gg = gbuf[bl*256 + j + 128];
        float go = gbuf[bl*256 + j + 192];
        float cc = creg[half][s];
        cc = sigf(gf)*cc + sigf(gi)*tanhf(gg);
        creg[half][s] = cc;
        hbuf[(half*32 + bl)*HID_ + j] = sigf(go)*tanhf(cc);
      }
      __syncthreads();
    }
  }
  // m0 / v0 / kl0
  for (int i = tid; i < B_SZ*DX_; i += 256) {
    int b = i >> 4, d = i & 15;
    float sm = bm[d], sv = bv[d];
    for (int k = 0; k < HID_; ++k) {
      float h = hbuf[b*HID_ + k];
      sm += h * Wm[d*HID_ + k];
      sv += h * Wv[d*HID_ + k];
    }
    float vv = (sv > 20.f ? sv : log1pf(__expf(sv))) + 1e-4f;
    m0[i] = sm; v0[i] = vv;
    float kl = 0.5f*(vv + sm*sm - 1.f - logf(vv)) / (float)B_SZ;
    atomicAdd(&accum[0], kl);
  }
}

// ---------------------------------------------------------------- GP precompute
__global__ __launch_bounds__(128) void k_gp(
    const float* __restrict__ Z, const float* __restrict__ log_ls,
    const float* __restrict__ log_os, const float* __restrict__ q_mu,
    const float* __restrict__ q_L, float* __restrict__ alpha,
    float* __restrict__ z2g, float* __restrict__ accum)
{
  __shared__ float Lt[(M_*(M_+1))/2];   // packed lower-tri 128x128 (33KB)
  __shared__ float z2s[M_];
  __shared__ float yv[M_], xv[M_];
  const int d = blockIdx.x;
  const int tid = threadIdx.x;
  const float* Zd = Z + d*M_*DX_;
  const float inv2 = __expf(-2.f*log_ls[d]);
  const float osc  = __expf(log_os[d]);

  {  // z2
    float s = 0.f;
    for (int i = 0; i < DX_; ++i) { float z = Zd[tid*DX_ + i]; s += z*z; }
    z2s[tid] = s; z2g[d*M_ + tid] = s;
  }
  __syncthreads();
  // build lower triangle of Kuu
  for (int k = 0; k <= tid; ++k) {
    float dotv = 0.f;
    for (int i = 0; i < DX_; ++i) dotv += Zd[tid*DX_+i]*Zd[k*DX_+i];
    float sq = z2s[tid] + z2s[k] - 2.f*dotv;
    float kv = osc*__expf(-0.5f*sq*inv2);
    if (k == tid) kv += JIT_;
    Lt[tri(tid,k)] = kv;
  }
  __syncthreads();
  // Cholesky (right-looking)
  for (int k = 0; k < M_; ++k) {
    if (tid == k) Lt[tri(k,k)] = sqrtf(Lt[tri(k,k)]);
    __syncthreads();
    if (tid > k) Lt[tri(tid,k)] /= Lt[tri(k,k)];
    __syncthreads();
    if (tid > k) {
      float lik = Lt[tri(tid,k)];
      for (int j = k+1; j <= tid; ++j) Lt[tri(tid,j)] -= lik*Lt[tri(j,k)];
    }
    __syncthreads();
  }
  // forward: L y = q_mu[d]
  yv[tid] = q_mu[d*M_ + tid];
  __syncthreads();
  for (int k = 0; k < M_; ++k) {
    if (tid == k) yv[k] /= Lt[tri(k,k)];
    __syncthreads();
    if (tid > k) yv[tid] -= Lt[tri(tid,k)]*yv[k];
    __syncthreads();
  }
  float bnorm = yv[tid]*yv[tid];
  // backward: L^T a = y
  xv[tid] = yv[tid];
  __syncthreads();
  for (int k = M_-1; k >= 0; --k) {
    if (tid == k) xv[k] /= Lt[tri(k,k)];
    __syncthreads();
    if (tid < k) xv[tid] -= Lt[tri(k,tid)]*xv[k];
    __syncthreads();
  }
  alpha[d*M_ + tid] = xv[tid];
  // ||L^{-1} q_L||_F^2 : one column per thread
  float tcol[M_];
  float afro = 0.f;
  for (int i = 0; i < M_; ++i) {
    float s = q_L[(d*M_ + i)*M_ + tid];
    for (int j = 0; j < i; ++j) s -= Lt[tri(i,j)]*tcol[j];
    float ti = s / Lt[tri(i,i)];
    tcol[i] = ti;
    afro += ti*ti;
  }
  float contrib = afro + bnorm
                + 2.f*logf(Lt[tri(tid,tid)])
                - 2.f*logf(fabsf(q_L[(d*M_ + tid)*M_ + tid]))
                - 1.f;
  atomicAdd(&accum[1], 0.5f*contrib / ((float)T_STEPS * (float)B_SZ));
}

// ---------------------------------------------------------------- x0 sampling
__global__ void k_sample(const float* __restrict__ m0, const float* __restrict__ v0,
                         const float* __restrict__ eps, float* __restrict__ x)
{
  int i = blockIdx.x*256 + threadIdx.x;     // over NMC*B*DX = 102400
  if (i < NPTS*DX_) {
    int bd = i % (B_SZ*DX_);
    x[i] = m0[bd] + sqrtf(v0[bd]) * eps[i];
  }
}

// ---------------------------------------------------------------- GP predict + RealNVP flow
__global__ __launch_bounds__(128) void k_predict_flow(
    const float* __restrict__ xcur, const float* __restrict__ Z,
    const float* __restrict__ log_ls, const float* __restrict__ log_os,
    const float* __restrict__ alpha, const float* __restrict__ z2g,
    const float* __restrict__ sW1, const float* __restrict__ sb1,
    const float* __restrict__ sW2, const float* __restrict__ sb2,
    const float* __restrict__ sW3, const float* __restrict__ sb3,
    const float* __restrict__ tW1, const float* __restrict__ tb1,
    const float* __restrict__ tW2, const float* __restrict__ tb2,
    const float* __restrict__ tW3, const float* __restrict__ tb3,
    float* __restrict__ xt)
{
  // 4 waves per block; each wave owns one 16-point row tile. grid=100 -> 400 tiles.
  __shared__ float sXT[4][16*16];
  __shared__ float sF [4][16*16];
  __shared__ float sH1[4][16*32];
  __shared__ float sH2[4][16*64];
  __shared__ float sS [4][16*16];
  __shared__ float sx2[4][16];
  const int wave = threadIdx.x >> 5;
  const int lane = threadIdx.x & 31;
  const int lo = lane & 15, hi = lane >> 4;
  const int tile = blockIdx.x*4 + wave;
  const int row0 = tile*16;
  float* XT = sXT[wave]; float* F = sF[wave];
  float* H1 = sH1[wave]; float* H2 = sH2[wave];
  float* Ss = sS[wave];  float* x2 = sx2[wave];

  for (int i = lane; i < 256; i += 32) XT[i] = xcur[row0*DX_ + i];
  if (hi == 0) {
    float s = 0.f;
    for (int dd = 0; dd < DX_; ++dd) { float v = XT[lo*DX_ + dd]; s += v*v; }
    x2[lo] = s;
  }
  // A fragments of the x-tile (K=16 -> 4 k-steps), reused for all 16 output dims
  v2f afr[4];
  #pragma unroll
  for (int kk = 0; kk < 4; ++kk) {
    int kb = kk*4 + 2*hi;
    afr[kk].x = XT[lo*DX_ + kb];
    afr[kk].y = XT[lo*DX_ + kb + 1];
  }
  // ---- GP predictive mean per output dim d ----
  for (int d = 0; d < DX_; ++d) {
    const float inv2 = __expf(-2.f*log_ls[d]);
    const float osc  = __expf(log_os[d]);
    const float* Zd  = Z + d*M_*DX_;
    const float* z2d = z2g + d*M_;
    const float* ad  = alpha + d*M_;
    float part[8];
    #pragma unroll
    for (int v = 0; v < 8; ++v) part[v] = 0.f;
    for (int nt = 0; nt < 8; ++nt) {
      v8f c = {0.f,0.f,0.f,0.f,0.f,0.f,0.f,0.f};
      #pragma unroll
      for (int kk = 0; kk < 4; ++kk) {
        int kb = kk*4 + 2*hi;
        v2f bb;                           // B(k=i, n=m) = Z[d][m][i]
        bb.x = Zd[(nt*16+lo)*DX_ + kb];
        bb.y = Zd[(nt*16+lo)*DX_ + kb + 1];
        c = wmma4(afr[kk], bb, c);
      }
      int j = nt*16 + lo;
      float z2j = z2d[j], aj = ad[j];
      #pragma unroll
      for (int v = 0; v < 8; ++v) {
        int p = v + 8*hi;
        float sq = x2[p] + z2j - 2.f*c[v];
        part[v] += aj * osc * __expf(-0.5f*sq*inv2);
      }
    }
    #pragma unroll
    for (int v = 0; v < 8; ++v) {       // reduce over the 16-lane (j) dimension
      float s = part[v];
      s += __shfl_xor(s, 1, 32);
      s += __shfl_xor(s, 2, 32);
      s += __shfl_xor(s, 4, 32);
      s += __shfl_xor(s, 8, 32);
      if (lo == 0) F[(v + 8*hi)*DX_ + d] = s;
    }
  }
  // ---- RealNVP flow: 4 coupling layers ----
  for (int layer = 0; layer < 4; ++layer) {
    for (int i = lane; i < 256; i += 32) {          // x_ = x * mk
      int dim = i & 15;
      bool mk = (layer & 1) ? (dim < 8) : (dim >= 8);
      XT[i] = mk ? F[i] : 0.f;
    }
    for (int net = 0; net < 2; ++net) {
      const float* W1 = (net ? tW1 : sW1) + layer*32*16;
      const float* B1 = (net ? tb1 : sb1) + layer*32;
      const float* W2 = (net ? tW2 : sW2) + layer*64*32;
      const float* B2 = (net ? tb2 : sb2) + layer*64;
      const float* W3 = (net ? tW3 : sW3) + layer*16*64;
      const float* B3 = (net ? tb3 : sb3) + layer*16;
      // GEMM1: [16x16] x W1^T -> [16x32], leaky
      for (int nt = 0; nt < 2; ++nt) {
        v8f c = {0.f,0.f,0.f,0.f,0.f,0.f,0.f,0.f};
        #pragma unroll
        for (int kk = 0; kk < 4; ++kk) {
          int kb = kk*4 + 2*hi;
          v2f a, bb;
          a.x  = XT[lo*16 + kb];              a.y  = XT[lo*16 + kb + 1];
          bb.x = W1[(nt*16+lo)*16 + kb];      bb.y = W1[(nt*16+lo)*16 + kb + 1];
          c = wmma4(a, bb, c);
        }
        float bias = B1[nt*16 + lo];
        #pragma unroll
        for (int v = 0; v < 8; ++v)
          H1[(v + 8*hi)*32 + nt*16 + lo] = leaky(c[v] + bias);
      }
      // GEMM2: [16x32] x W2^T -> [16x64], leaky
      for (int nt = 0; nt < 4; ++nt) {
        v8f c = {0.f,0.f,0.f,0.f,0.f,0.f,0.f,0.f};
        #pragma unroll
        for (int kk = 0; kk < 8; ++kk) {
          int kb = kk*4 + 2*hi;
          v2f a, bb;
          a.x  = H1[lo*32 + kb];              a.y  = H1[lo*32 + kb + 1];
          bb.x = W2[(nt*16+lo)*32 + kb];      bb.y = W2[(nt*16+lo)*32 + kb + 1];
          c = wmma4(a, bb, c);
        }
        float bias = B2[nt*16 + lo];
        #pragma unroll
        for (int v = 0; v < 8; ++v)
          H2[(v + 8*hi)*64 + nt*16 + lo] = leaky(c[v] + bias);
      }
      // GEMM3: [16x64] x W3^T -> [16x16]
      v8f c3 = {0.f,0.f,0.f,0.f,0.f,0.f,0.f,0.f};
      #pragma unroll
      for (int kk = 0; kk < 16; ++kk) {
        int kb = kk*4 + 2*hi;
        v2f a, bb;
        a.x  = H2[lo*64 + kb];                a.y  = H2[lo*64 + kb + 1];
        bb.x = W3[lo*64 + kb];                bb.y = W3[lo*64 + kb + 1];
        c3 = wmma4(a, bb, c3);
      }
      float bias3 = B3[lo];
      bool notmk = (layer & 1) ? (lo >= 8) : (lo < 8);    // (1-mk) on output dim
      if (net == 0) {
        #pragma unroll
        for (int v = 0; v < 8; ++v)
          Ss[(v + 8*hi)*16 + lo] = notmk ? (c3[v] + bias3) : 0.f;
      } else {
        #pragma unroll
        for (int v = 0; v < 8; ++v) {
          int idx = (v + 8*hi)*16 + lo;
          float tval = c3[v] + bias3;
          float sval = Ss[idx];
          float xold = F[idx];
          float xm   = XT[idx];
          F[idx] = xm + (notmk ? (xold*__expf(sval) + tval) : 0.f);
        }
      }
    }
  }
  for (int i = lane; i < 256; i += 32) xt[row0*DX_ + i] = F[i];
}

// ---------------------------------------------------------------- EnKF update (one block per batch)
__global__ __launch_bounds__(256) void k_enkf(
    const float* __restrict__ xt, const float* __restrict__ obs,
    const float* __restrict__ Hm, int t,
    float* __restrict__ xcur, float* __restrict__ means_out,
    float* __restrict__ pvars_out, float* __restrict__ accum)
{
  __shared__ float Xl[NMC_*DX_];    // prior ensemble for this b
  __shared__ float Inn[NMC_*DY_];
  __shared__ float Xp[NMC_*DX_];
  __shared__ float Xm[DX_];
  __shared__ float Pm[256], Sm[256], Kt[256], KH[256], Hl[256];
  const int b = blockIdx.x;
  const int tid = threadIdx.x;

  for (int i = tid; i < NMC_*DX_; i += 256) {
    int n = i >> 4, dd = i & 15;
    Xl[i] = xt[(n*B_SZ + b)*DX_ + dd];
  }
  Hl[tid] = Hm[tid];
  __syncthreads();
  if (tid < DX_) {
    float s = 0.f;
    for (int n = 0; n < NMC_; ++n) s += Xl[n*DX_ + tid];
    Xm[tid] = s / (float)NMC_;
  }
  __syncthreads();
  { // localized covariance P = taper * cov + PVAR*I
    int dd = tid >> 4, e = tid & 15;
    float s = 0.f;
    for (int n = 0; n < NMC_; ++n)
      s += (Xl[n*DX_ + dd] - Xm[dd]) * (Xl[n*DX_ + e] - Xm[e]);
    Pm[tid] = gc_taper(dd, e) * s/(float)(NMC_-1) + (dd == e ? PVAR_ : 0.f);
  }
  __syncthreads();
  { // S = H P H^T + R
    int o = tid >> 4, q = tid & 15;
    float s = 0.f;
    for (int dd = 0; dd < DX_; ++dd) {
      float hod = Hl[o*16 + dd];
      for (int e = 0; e < DX_; ++e) s += hod * Pm[dd*16 + e] * Hl[q*16 + e];
    }
    Sm[tid] = s + (o == q ? EVAR_ : 0.f);
  }
  __syncthreads();
  // Cholesky of S (16x16, lower in place)
  for (int k = 0; k < DY_; ++k) {
    if (tid == 0) Sm[k*16+k] = sqrtf(Sm[k*16+k]);
    __syncthreads();
    if (tid > k && tid < 16) Sm[tid*16+k] /= Sm[k*16+k];
    __syncthreads();
    if (tid > k && tid < 16) {
      float lik = Sm[tid*16+k];
      for (int j = k+1; j <= tid; ++j) Sm[tid*16+j] -= lik*Sm[j*16+k];
    }
    __syncthreads();
  }
  // K^T: solve S Kt = (P H^T)^T, one column per thread (columns over DX)
  if (tid < DX_) {
    int dcol = tid;
    float tv[16];
    for (int i = 0; i < DY_; ++i) {
      float s = 0.f;
      for (int e = 0; e < DX_; ++e) s += Pm[dcol*16 + e] * Hl[i*16 + e]; // rhs[i][dcol]
      for (int j = 0; j < i; ++j) s -= Sm[i*16+j]*tv[j];
      tv[i] = s / Sm[i*16+i];
    }
    for (int i = DY_-1; i >= 0; --i) {
      float s = tv[i];
      for (int j = i+1; j < DY_; ++j) s -= Sm[j*16+i]*Kt[j*16 + dcol];
      Kt[i*16 + dcol] = s / Sm[i*16+i];
    }
  }
  __syncthreads();
  { // KH[d][e] = sum_o Kt[o][d] H[o][e]
    int dd = tid >> 4, e = tid & 15;
    float s = 0.f;
    for (int o = 0; o < DY_; ++o) s += Kt[o*16 + dd] * Hl[o*16 + e];
    KH[tid] = s;
  }
  __syncthreads();
  if (tid < DX_) { // pvars: diag((I-KH) P)
    float s = 0.f;
    for (int e = 0; e < DX_; ++e) {
      float ike = (tid == e ? 1.f : 0.f) - KH[tid*16 + e];
      s += ike * Pm[e*16 + tid];
    }
    pvars_out[(b*T_STEPS + t)*DX_ + tid] = s;
  }
  // innovations
  for (int i = tid; i < NMC_*DY_; i += 256) {
    int n = i >> 4, o = i & 15;
    float s = obs[(b*T_STEPS + t)*DY_ + o];
    for (int dd = 0; dd < DX_; ++dd) s -= Xl[n*DX_ + dd] * Hl[o*16 + dd];
    Inn[i] = s;
  }
  __syncthreads();
  // posterior ensemble -> back to global state
  for (int i = tid; i < NMC_*DX_; i += 256) {
    int n = i >> 4, dd = i & 15;
    float s = Xl[i];
    for (int o = 0; o < DY_; ++o) s += Inn[n*16 + o] * Kt[o*16 + dd];
    Xp[i] = s;
    xcur[(n*B_SZ + b)*DX_ + dd] = s;
  }
  __syncthreads();
  if (tid < DX_) {
    float s = 0.f;
    for (int n = 0; n < NMC_; ++n) s += Xp[n*DX_ + tid];
    means_out[(b*T_STEPS + t)*DX_ + tid] = s / (float)NMC_;
  }
  if (tid == 0) { // marginal log-likelihood for this (b,t)
    float dv[16], tv[16], sv[16];
    for (int o = 0; o < DY_; ++o) {
      float s = obs[(b*T_STEPS + t)*DY_ + o];
      for (int dd = 0; dd < DX_; ++dd) s -= Xm[dd] * Hl[o*16 + dd];
      dv[o] = s;
    }
    for (int i = 0; i < DY_; ++i) {
      float s = dv[i];
      for (int j = 0; j < i; ++j) s -= Sm[i*16+j]*tv[j];
      tv[i] = s / Sm[i*16+i];
    }
    for (int i = DY_-1; i >= 0; --i) {
      float s = tv[i];
      for (int j = i+1; j < DY_; ++j) s -= Sm[j*16+i]*sv[j];
      sv[i] = s / Sm[i*16+i];
    }
    float quad = 0.f, ld = 0.f;
    for (int o = 0; o < DY_; ++o) { quad += dv[o]*sv[o]; ld += logf(Sm[o*16+o]); }
    atomicAdd(&accum[2], -0.5f*(quad + 2.f*ld + (float)DY_*LOG2PI_));
  }
}

// ---------------------------------------------------------------- finalize
__global__ void k_final(const float* __restrict__ accum, float* __restrict__ out) {
  if (threadIdx.x == 0 && blockIdx.x == 0)
    out[0] = -accum[0] - accum[1] + accum[2] / (float)T_STEPS;
}

extern "C" void kernel_launch(void* const* d_in, const int* in_sizes, int n_in,
                              void* d_out, int out_size, void* d_ws, size_t ws_size,
                              hipStream_t stream) {
  const float* obs    = (const float*)d_in[0];
  const float* eps    = (const float*)d_in[1];
  const float* Wih    = (const float*)d_in[2];
  const float* Whh    = (const float*)d_in[3];
  const float* b_lstm = (const float*)d_in[4];
  const float* Wm     = (const float*)d_in[5];
  const float* bm     = (const float*)d_in[6];
  const float* Wv     = (const float*)d_in[7];
  const float* bv     = (const float*)d_in[8];
  const float* Z      = (const float*)d_in[9];
  const float* log_ls = (const float*)d_in[10];
  const float* log_os = (const float*)d_in[11];
  const float* q_mu   = (const float*)d_in[12];
  const float* q_L    = (const float*)d_in[13];
  const float* sW1 = (const float*)d_in[14]; const float* sb1 = (const float*)d_in[15];
  const float* sW2 = (const float*)d_in[16]; const float* sb2 = (const float*)d_in[17];
  const float* sW3 = (const float*)d_in[18]; const float* sb3 = (const float*)d_in[19];
  const float* tW1 = (const float*)d_in[20]; const float* tb1 = (const float*)d_in[21];
  const float* tW2 = (const float*)d_in[22]; const float* tb2 = (const float*)d_in[23];
  const float* tW3 = (const float*)d_in[24]; const float* tb3 = (const float*)d_in[25];
  const float* Hm  = (const float*)d_in[26];

  float* out = (float*)d_out;
  float* ws  = (float*)d_ws;
  float* accum = ws;                   // 4
  float* m0    = ws + 4;               // 1024
  float* v0    = m0 + B_SZ*DX_;        // 1024
  float* alpha = v0 + B_SZ*DX_;        // 2048
  float* z2    = alpha + DX_*M_;       // 2048
  float* xcur  = z2 + DX_*M_;          // 102400
  float* xtb   = xcur + NPTS*DX_;      // 102400

  float* means = out + 1;
  float* pvars = out + 1 + B_SZ*T_STEPS*DX_;

  k_zero<<<1, 64, 0, stream>>>(accum);
  k_lstm<<<1, 256, 0, stream>>>(obs, Wih, Whh, b_lstm, Wm, bm, Wv, bv, m0, v0, accum);
  k_gp<<<16, 128, 0, stream>>>(Z, log_ls, log_os, q_mu, q_L, alpha, z2, accum);
  k_sample<<<(NPTS*DX_ + 255)/256, 256, 0, stream>>>(m0, v0, eps, xcur);
  for (int t = 0; t < T_STEPS; ++t) {
    k_predict_flow<<<NPTS/64, 128, 0, stream>>>(xcur, Z, log_ls, log_os, alpha, z2,
        sW1, sb1, sW2, sb2, sW3, sb3, tW1, tb1, tW2, tb2, tW3, tb3, xtb);
    k_enkf<<<B_SZ, 256, 0, stream>>>(xtb, obs, Hm, t, xcur, means, pvars, accum);
  }
  k_final<<<1, 64, 0, stream>>>(accum, out);
}